// DifferentiableEnvelopeFollower_28810640621777
// MI455X (gfx1250) — compile-verified
//
#include <hip/hip_runtime.h>
#include <stdint.h>

// ---------------------------------------------------------------------------
// DifferentiableEnvelopeFollower for MI455X (gfx1250, wave32)
//
// Serial nonlinear IIR per row: latency-bound on a 3-VALU dependent chain.
// Memory is staged through LDS with gfx1250 async global->LDS DMA (ASYNCcnt)
// + global_prefetch so the 246MB of traffic hides entirely under the scan.
// ---------------------------------------------------------------------------

#define ROWS    32          // rows per block == lanes per wave
#define TILE    128         // samples per tile (TILE*4 = 512B = 32 lanes * 16B)
#define PADF    4           // LDS row padding (floats), keeps 16B alignment
#define STRIDEF (TILE + PADF)

typedef __attribute__((ext_vector_type(4))) int int4v;   // 16-byte payload type

#if defined(__HIP_DEVICE_COMPILE__)
  #if __has_builtin(__builtin_amdgcn_global_load_async_to_lds_b128)
    #define HAVE_ASYNC_LDS 1
  #else
    #define HAVE_ASYNC_LDS 0
  #endif
  #if __has_builtin(__builtin_amdgcn_s_wait_asynccnt)
    #define HAVE_WAIT_ASYNC 1
  #else
    #define HAVE_WAIT_ASYNC 0
  #endif
#else
  #define HAVE_ASYNC_LDS 0
  #define HAVE_WAIT_ASYNC 0
#endif

template <int N>
__device__ __forceinline__ void wait_asynccnt_le() {
#if HAVE_WAIT_ASYNC
  __builtin_amdgcn_s_wait_asynccnt(N);
  asm volatile("" ::: "memory");
#elif defined(__HIP_DEVICE_COMPILE__) && HAVE_ASYNC_LDS
  asm volatile("s_wait_asynccnt %0" :: "n"(N) : "memory");
#else
  asm volatile("" ::: "memory");
#endif
}

__device__ __forceinline__ float env_step(float env, float xv, float ac, float rc) {
  float lv = fabsf(xv);
  float c  = (lv > env) ? ac : rc;          // v_cmp + v_cndmask
  return fmaf(c, env - lv, lv);             // c*env + (1-c)*lv  (v_sub + v_fma)
}

__global__ __launch_bounds__(ROWS)
void envfollow_kernel(const float* __restrict__ x,
                      const float* __restrict__ p_raw_attack,
                      const float* __restrict__ p_raw_release,
                      const int*   __restrict__ p_sample_rate,
                      float* __restrict__ out,
                      int L)
{
  __shared__ __align__(16) float s_in[2][ROWS][STRIDEF];   // double-buffered input
  __shared__ __align__(16) float s_out[ROWS][STRIDEF];     // staged output tile

  const int  lane    = threadIdx.x;                // 0..31, one row per lane
  const int  rowBase = blockIdx.x * ROWS;
  const long rowOff  = (long)(rowBase + lane) * (long)L;

  // --- coefficients (computed once per thread, matches the JAX reference) ---
  const float sr     = (float)p_sample_rate[0];
  const float att_ms = 0.1f + 49.9f  / (1.0f + expf(-p_raw_attack[0]));
  const float rel_ms = 10.0f + 490.0f / (1.0f + expf(-p_raw_release[0]));
  const float ac     = expf(-1000.0f / (att_ms * sr));   // attack coeff
  const float rc     = expf(-1000.0f / (rel_ms * sr));   // release coeff

  // Issue one tile's worth of loads: 32 async B128 DMAs, each moving 512
  // contiguous bytes of one row into its padded LDS row (fully coalesced).
  auto issue_tile = [&](int t0, int buf) {
#if HAVE_ASYNC_LDS
    #pragma unroll 4
    for (int r = 0; r < ROWS; ++r) {
      const float* g = x + (long)(rowBase + r) * (long)L + t0 + lane * 4;
      float*       l = &s_in[buf][r][lane * 4];
      __builtin_amdgcn_global_load_async_to_lds_b128(
          (__attribute__((address_space(1))) int4v*)(uintptr_t)g,
          (__attribute__((address_space(3))) int4v*)l,
          /*imm offset*/0, /*cpol*/0);
    }
#else
    #pragma unroll 4
    for (int r = 0; r < ROWS; ++r) {
      const float4 v = *(const float4*)(x + (long)(rowBase + r) * (long)L + t0 + lane * 4);
      *(float4*)&s_in[buf][r][lane * 4] = v;
    }
#endif
  };

  float env = 0.0f;

  // Fast path requires 16B-aligned tiles in global memory.
  int nFull = ((L & 3) == 0) ? (L / TILE) : 0;

  if (nFull > 0) issue_tile(0, 0);

  for (int t = 0; t < nFull; ++t) {
    const int  buf  = t & 1;
    const int  t0   = t * TILE;
    const bool more = (t + 1) < nFull;

    // Keep one tile of DMA in flight while we scan the current one.
    if (more) issue_tile(t0 + TILE, buf ^ 1);
    if (more) wait_asynccnt_le<ROWS>();   // in-order: current tile's 32 DMAs done
    else      wait_asynccnt_le<0>();

    // Pull tile t+2 toward L2 so the t+1 DMAs are short-latency.
    if (t + 2 < nFull) {
      const int pr = lane >> 2;                   // 8 rows per prefetch instr
      const int pc = (lane & 3) * 32;             // 4 x 128B cachelines per row
      const float* pf = x + (long)(rowBase + pr) * (long)L + (t0 + 2 * TILE) + pc;
      #pragma unroll
      for (int g8 = 0; g8 < ROWS; g8 += 8)
        __builtin_prefetch(pf + (long)g8 * (long)L, 0, 0);
    }

    // Serial scan of this lane's row tile: 1 ds_load_b128 + 1 ds_store_b128
    // per 4 steps; dependent chain is 3 VALU ops per step.
    for (int k = 0; k < TILE; k += 4) {
      const float4 xv = *(const float4*)&s_in[buf][lane][k];
      float4 ev;
      env = env_step(env, xv.x, ac, rc); ev.x = env;
      env = env_step(env, xv.y, ac, rc); ev.y = env;
      env = env_step(env, xv.z, ac, rc); ev.z = env;
      env = env_step(env, xv.w, ac, rc); ev.w = env;
      *(float4*)&s_out[lane][k] = ev;
    }

    // Cooperative, coalesced store: 512B contiguous per row per instruction.
    #pragma unroll 4
    for (int r = 0; r < ROWS; ++r) {
      const float4 v = *(const float4*)&s_out[r][lane * 4];
      *(float4*)(out + (long)(rowBase + r) * (long)L + t0 + lane * 4) = v;
    }
  }

  // Scalar tail (also the whole-L fallback if L wasn't float4-aligned).
  for (int k = nFull * TILE; k < L; ++k) {
    env = env_step(env, x[rowOff + k], ac, rc);
    out[rowOff + k] = env;
  }
}

extern "C" void kernel_launch(void* const* d_in, const int* in_sizes, int n_in,
                              void* d_out, int out_size, void* d_ws, size_t ws_size,
                              hipStream_t stream) {
  (void)d_ws; (void)ws_size; (void)n_in; (void)out_size;
  const float* x  = (const float*)d_in[0];
  const float* ra = (const float*)d_in[1];
  const float* rr = (const float*)d_in[2];
  const int*   sr = (const int*)d_in[3];
  float*       o  = (float*)d_out;

  const int B = 64;                      // fixed by the reference setup
  const int L = in_sizes[0] / B;         // 480000

  dim3 grid(B / ROWS), block(ROWS);      // 2 blocks x 1 wave32, lane == row
  envfollow_kernel<<<grid, block, 0, stream>>>(x, ra, rr, sr, o, L);
}